// ProtoPGNNNet_22514218566446
// MI455X (gfx1250) — compile-verified
//
#include <hip/hip_runtime.h>
#include <math.h>

typedef __attribute__((ext_vector_type(2))) float v2f;
typedef __attribute__((ext_vector_type(8))) float v8f;

#define N_NODES 262144
#define N_EDGES 2097152
#define NGRAPH  2048
#define HIDDIM  128
#define NPROTO  20
#define BN_EPS  1e-5f
#define NCHUNK  1024          // scan chunks: 1024 chunks x 256 nodes

// ---------------------------------------------------------------- utilities
__global__ void fill_zero_u32(unsigned* __restrict__ p, int n) {
  int i = blockIdx.x * blockDim.x + threadIdx.x;
  if (i < n) p[i] = 0u;
}

// ---------------------------------------------------------------- CSR build
__global__ void hist_kernel(const int* __restrict__ dst, int* __restrict__ degi) {
  int e = blockIdx.x * blockDim.x + threadIdx.x;
  if (e < N_EDGES) atomicAdd(&degi[dst[e]], 1);
}

// per-chunk sums (1024 threads total, each sums 256 counts)
__global__ void chunk_sum_kernel(const int* __restrict__ degi, int* __restrict__ bsum) {
  int t = blockIdx.x * blockDim.x + threadIdx.x;
  if (t >= NCHUNK) return;
  int s = 0;
  const int base = t * 256;
  for (int i = 0; i < 256; ++i) s += degi[base + i];
  bsum[t] = s;
}

// serial exclusive scan of the 1024 chunk sums (tiny, deterministic)
__global__ void chunk_scan_kernel(const int* __restrict__ bsum, int* __restrict__ bbase) {
  if (blockIdx.x == 0 && threadIdx.x == 0) {
    int run = 0;
    for (int t = 0; t < NCHUNK; ++t) { bbase[t] = run; run += bsum[t]; }
  }
}

// per-chunk exclusive scan -> node offsets (+ scatter cursors)
__global__ void offsets_kernel(const int* __restrict__ degi, const int* __restrict__ bbase,
                               int* __restrict__ off, int* __restrict__ cursor) {
  int t = blockIdx.x * blockDim.x + threadIdx.x;
  if (t >= NCHUNK) return;
  int run = bbase[t];
  const int base = t * 256;
  for (int i = 0; i < 256; ++i) {
    off[base + i] = run;
    cursor[base + i] = run;
    run += degi[base + i];
  }
}

__global__ void scatter_kernel(const int* __restrict__ src, const int* __restrict__ dst,
                               int* __restrict__ cursor, int* __restrict__ csr) {
  int e = blockIdx.x * blockDim.x + threadIdx.x;
  if (e >= N_EDGES) return;
  int pos = atomicAdd(&cursor[dst[e]], 1);
  csr[pos] = src[e];
}

// ---------------------------------------------------------------- pull aggregation
// one wave per node; lane owns 4 consecutive floats (float4), mean folded in; no atomics
__global__ __launch_bounds__(256) void agg_kernel(
    const float* __restrict__ H, const int* __restrict__ csr,
    const int* __restrict__ off, const int* __restrict__ degi,
    float* __restrict__ NB) {
  int gt = blockIdx.x * blockDim.x + threadIdx.x;
  int node = gt >> 5, lane = gt & 31;
  if (node >= N_NODES) return;
  const int beg = off[node], dg = degi[node];
  float4 acc = make_float4(0.f, 0.f, 0.f, 0.f);
  for (int e = beg; e < beg + dg; ++e) {
    int s = csr[e];
    const float4 v = *(const float4*)(H + (size_t)s * HIDDIM + lane * 4);
    acc.x += v.x; acc.y += v.y; acc.z += v.z; acc.w += v.w;
  }
  const float sc = 1.0f / fmaxf((float)dg, 1.0f);
  acc.x *= sc; acc.y *= sc; acc.z *= sc; acc.w *= sc;
  *(float4*)(NB + (size_t)node * HIDDIM + lane * 4) = acc;
}

// ---------------------------------------------------------------- WMMA GEMM helpers
// Weights staged in LDS k-pair interleaved: lw[kp*256 + col*2 + j] = W[2*kp+j][col]
// -> each B fragment is a single contiguous ds_load_b64 into an aligned VGPR pair.
__device__ __forceinline__ void stage_weights(const float* __restrict__ W, float* lw, int tid) {
  for (int idx = tid; idx < 64 * HIDDIM; idx += 256) {
    const int kp = idx >> 7, col = idx & 127;
    v2f w2 = { W[(2 * kp) * HIDDIM + col], W[(2 * kp + 1) * HIDDIM + col] };
    *(v2f*)(lw + kp * 256 + col * 2) = w2;
  }
}

// C[N,128] = A[N,128] @ W[128,128] + bias   (FP32 WMMA 16x16x4)
__global__ __launch_bounds__(256) void gemm_embed_kernel(
    const float* __restrict__ A, const float* __restrict__ W,
    const float* __restrict__ bias, float* __restrict__ C) {
  extern __shared__ float lw[];  // 64*256 floats = 64 KB
  const int tid = threadIdx.x;
  stage_weights(W, lw, tid);
  __syncthreads();

  const int wave = tid >> 5, lane = tid & 31;
  const int hi = lane >> 4, nn = lane & 15;
  const int r0 = blockIdx.x * 128 + wave * 16;
  const int m = r0 + nn;

  v8f acc[8];
#pragma unroll
  for (int t = 0; t < 8; ++t) {          // bias folded into accumulator init
    const float bv = bias[t * 16 + nn];
#pragma unroll
    for (int v = 0; v < 8; ++v) acc[t][v] = bv;
  }
#pragma unroll 4
  for (int ks = 0; ks < 32; ++ks) {
    const int k0 = 4 * ks + 2 * hi;
    const int kp = 2 * ks + hi;
    v2f a = *(const v2f*)(A + (size_t)m * HIDDIM + k0);
#pragma unroll
    for (int t = 0; t < 8; ++t) {
      v2f b = *(const v2f*)(lw + kp * 256 + (t * 16 + nn) * 2);
      acc[t] = __builtin_amdgcn_wmma_f32_16x16x4_f32(
          false, a, false, b, (short)0, acc[t], false, false);
    }
  }
#pragma unroll
  for (int t = 0; t < 8; ++t)
#pragma unroll
    for (int v = 0; v < 8; ++v)
      C[(size_t)(r0 + v + 8 * hi) * HIDDIM + t * 16 + nn] = acc[t][v];
}

// act = relu(H @ Wself + NB @ Wneigh + bias), written in-place over NB
__global__ __launch_bounds__(256) void gemm_conv_kernel(
    const float* __restrict__ H, float* __restrict__ NB,
    const float* __restrict__ Wself, const float* __restrict__ Wneigh,
    const float* __restrict__ bias) {
  extern __shared__ float lw[];  // 2 * 64*256 floats = 128 KB
  float* lws = lw;
  float* lwn = lw + 64 * 256;
  const int tid = threadIdx.x;
  stage_weights(Wself, lws, tid);
  stage_weights(Wneigh, lwn, tid);
  __syncthreads();

  const int wave = tid >> 5, lane = tid & 31;
  const int hi = lane >> 4, nn = lane & 15;
  const int r0 = blockIdx.x * 128 + wave * 16;
  const int m = r0 + nn;

  v8f acc[8];
#pragma unroll
  for (int t = 0; t < 8; ++t) {
    const float bv = bias[t * 16 + nn];
#pragma unroll
    for (int v = 0; v < 8; ++v) acc[t][v] = bv;
  }
  // self path
#pragma unroll 4
  for (int ks = 0; ks < 32; ++ks) {
    const int k0 = 4 * ks + 2 * hi;
    const int kp = 2 * ks + hi;
    v2f a = *(const v2f*)(H + (size_t)m * HIDDIM + k0);
#pragma unroll
    for (int t = 0; t < 8; ++t) {
      v2f b = *(const v2f*)(lws + kp * 256 + (t * 16 + nn) * 2);
      acc[t] = __builtin_amdgcn_wmma_f32_16x16x4_f32(
          false, a, false, b, (short)0, acc[t], false, false);
    }
  }
  // neighbor path (NB already holds the mean)
#pragma unroll 4
  for (int ks = 0; ks < 32; ++ks) {
    const int k0 = 4 * ks + 2 * hi;
    const int kp = 2 * ks + hi;
    v2f a = *(const v2f*)(NB + (size_t)m * HIDDIM + k0);
#pragma unroll
    for (int t = 0; t < 8; ++t) {
      v2f b = *(const v2f*)(lwn + kp * 256 + (t * 16 + nn) * 2);
      acc[t] = __builtin_amdgcn_wmma_f32_16x16x4_f32(
          false, a, false, b, (short)0, acc[t], false, false);
    }
  }
  // relu, store act in place over NB (wave only reads/writes its own 16 rows)
#pragma unroll
  for (int t = 0; t < 8; ++t)
#pragma unroll
    for (int v = 0; v < 8; ++v)
      NB[(size_t)(r0 + v + 8 * hi) * HIDDIM + t * 16 + nn] = fmaxf(acc[t][v], 0.0f);
}

// ---------------------------------------------------------------- batch-norm
__global__ __launch_bounds__(128) void bn_stats_kernel(
    const float* __restrict__ act, float* __restrict__ sums, float* __restrict__ sumsq) {
  const int c = threadIdx.x;
  const int row0 = blockIdx.x * 256;
  float s = 0.0f, s2 = 0.0f;
  for (int r = 0; r < 256; ++r) {
    float v = act[(size_t)(row0 + r) * HIDDIM + c];
    s += v; s2 += v * v;
  }
  atomicAdd(&sums[c], s);
  atomicAdd(&sumsq[c], s2);
}

__global__ __launch_bounds__(256) void bn_apply_kernel(
    float* __restrict__ H, const float* __restrict__ act,
    const float* __restrict__ sums, const float* __restrict__ sumsq,
    const float* __restrict__ gamma, const float* __restrict__ beta) {
  int i = blockIdx.x * blockDim.x + threadIdx.x;
  const int c = i & (HIDDIM - 1);
  const float inv_n = 1.0f / (float)N_NODES;
  float mu = sums[c] * inv_n;
  float var = sumsq[c] * inv_n - mu * mu;
  float xh = (act[i] - mu) * rsqrtf(var + BN_EPS);
  H[i] = H[i] + gamma[c] * xh + beta[c];   // residual + BN
}

// ---------------------------------------------------------------- prototype head
// one wave per node; sim > 0 always, so int-bitpattern atomicMax with 0-init is valid
__global__ __launch_bounds__(256) void proto_sim_kernel(
    const float* __restrict__ H, const float* __restrict__ ppos,
    const float* __restrict__ pneg, const int* __restrict__ gid,
    float* __restrict__ ss) {
  int gt = blockIdx.x * blockDim.x + threadIdx.x;
  int node = gt >> 5, lane = gt & 31;
  if (node >= N_NODES) return;
  float hv[4];
  float hh = 0.0f;
#pragma unroll
  for (int q = 0; q < 4; ++q) {
    hv[q] = H[(size_t)node * HIDDIM + lane + 32 * q];
    hh += hv[q] * hv[q];
  }
#pragma unroll
  for (int o = 16; o > 0; o >>= 1) hh += __shfl_xor(hh, o, 32);
  const int g = gid[node];
  for (int j = 0; j < NPROTO; ++j) {
    const float* p = (j < 10) ? (ppos + (size_t)j * HIDDIM) : (pneg + (size_t)(j - 10) * HIDDIM);
    float c = 0.0f;
#pragma unroll
    for (int q = 0; q < 4; ++q) {
      float pv = p[lane + 32 * q];
      c += pv * pv - 2.0f * hv[q] * pv;
    }
#pragma unroll
    for (int o = 16; o > 0; o >>= 1) c += __shfl_xor(c, o, 32);
    if (lane == 0) {
      float d2 = fmaxf(hh + c, 0.0f);
      float sim = logf((d2 + 1.0f) / (d2 + 1e-12f));
      atomicMax((int*)&ss[g * NPROTO + j], __float_as_int(sim));
    }
  }
}

__global__ void head_kernel(const float* __restrict__ ss, const float* __restrict__ fcw,
                            float* __restrict__ y) {
  int b = blockIdx.x * blockDim.x + threadIdx.x;
  if (b >= NGRAPH) return;
  float s = 0.0f;
#pragma unroll
  for (int j = 0; j < NPROTO; ++j) s += ss[b * NPROTO + j] * fcw[j];
  y[b] = 1.0f / (1.0f + expf(-s));
}

// ---------------------------------------------------------------- launcher
extern "C" void kernel_launch(void* const* d_in, const int* in_sizes, int n_in,
                              void* d_out, int out_size, void* d_ws, size_t ws_size,
                              hipStream_t stream) {
  const float* x       = (const float*)d_in[0];   // [N,128]
  const float* W_embed = (const float*)d_in[1];   // [128,128]
  const float* b_embed = (const float*)d_in[2];   // [128]
  const float* W_self  = (const float*)d_in[3];   // [3,128,128]
  const float* W_neigh = (const float*)d_in[4];   // [3,128,128]
  const float* b_conv  = (const float*)d_in[5];   // [3,128]
  const float* gamma   = (const float*)d_in[6];   // [3,128]
  const float* beta    = (const float*)d_in[7];   // [3,128]
  const float* p_pos   = (const float*)d_in[8];   // [10,128]
  const float* p_neg   = (const float*)d_in[9];   // [10,128]
  const float* fc_w    = (const float*)d_in[10];  // [20]
  const int*   src     = (const int*)d_in[11];    // [E]
  const int*   dst     = (const int*)d_in[12];    // [E]
  const int*   gid     = (const int*)d_in[13];    // [N]
  float*       y       = (float*)d_out;           // [B]

  // workspace layout
  const size_t NH = (size_t)N_NODES * HIDDIM;
  float* d_h     = (float*)d_ws;             // N*128
  float* d_nb    = d_h + NH;                 // N*128 (neigh mean, then act in place)
  float* d_sums  = d_nb + NH;                // 128
  float* d_sumsq = d_sums + HIDDIM;          // 128
  float* d_ss    = d_sumsq + HIDDIM;         // B*20
  int*   d_degi  = (int*)(d_ss + NGRAPH * NPROTO);  // N
  int*   d_off   = d_degi + N_NODES;                // N
  int*   d_cur   = d_off + N_NODES;                 // N
  int*   d_csr   = d_cur + N_NODES;                 // E
  int*   d_bsum  = d_csr + N_EDGES;                 // 1024
  int*   d_bbase = d_bsum + NCHUNK;                 // 1024

  // ---- CSR build (once per call)
  fill_zero_u32<<<(N_NODES + 255) / 256, 256, 0, stream>>>((unsigned*)d_degi, N_NODES);
  fill_zero_u32<<<(NGRAPH * NPROTO + 255) / 256, 256, 0, stream>>>((unsigned*)d_ss,
                                                                   NGRAPH * NPROTO);
  hist_kernel<<<(N_EDGES + 255) / 256, 256, 0, stream>>>(dst, d_degi);
  chunk_sum_kernel<<<NCHUNK / 256, 256, 0, stream>>>(d_degi, d_bsum);
  chunk_scan_kernel<<<1, 1, 0, stream>>>(d_bsum, d_bbase);
  offsets_kernel<<<NCHUNK / 256, 256, 0, stream>>>(d_degi, d_bbase, d_off, d_cur);
  scatter_kernel<<<(N_EDGES + 255) / 256, 256, 0, stream>>>(src, dst, d_cur, d_csr);

  // ---- input embedding (WMMA GEMM)
  gemm_embed_kernel<<<N_NODES / 128, 256, 64 * 256 * sizeof(float), stream>>>(
      x, W_embed, b_embed, d_h);

  for (int l = 0; l < 3; ++l) {
    agg_kernel<<<(N_NODES * 32) / 256, 256, 0, stream>>>(d_h, d_csr, d_off, d_degi, d_nb);
    fill_zero_u32<<<1, 256, 0, stream>>>((unsigned*)d_sums, 2 * HIDDIM);
    gemm_conv_kernel<<<N_NODES / 128, 256, 2 * 64 * 256 * sizeof(float), stream>>>(
        d_h, d_nb, W_self + (size_t)l * HIDDIM * HIDDIM,
        W_neigh + (size_t)l * HIDDIM * HIDDIM, b_conv + l * HIDDIM);
    bn_stats_kernel<<<N_NODES / 256, 128, 0, stream>>>(d_nb, d_sums, d_sumsq);
    bn_apply_kernel<<<(int)(NH / 256), 256, 0, stream>>>(
        d_h, d_nb, d_sums, d_sumsq, gamma + l * HIDDIM, beta + l * HIDDIM);
  }

  // ---- prototype similarities + per-graph segment-max + sigmoid head
  proto_sim_kernel<<<(N_NODES * 32) / 256, 256, 0, stream>>>(d_h, p_pos, p_neg, gid, d_ss);
  head_kernel<<<(NGRAPH + 255) / 256, 256, 0, stream>>>(d_ss, fc_w, y);
}